// MLPDecoder_70549132804178
// MI455X (gfx1250) — compile-verified
//
#include <hip/hip_runtime.h>
#include <hip/hip_bf16.h>

typedef __attribute__((ext_vector_type(2))) float v2f;
typedef __attribute__((ext_vector_type(4))) float v4f;
typedef __attribute__((ext_vector_type(8))) float v8f;

#define DFEAT   64
#define HID     128
#define TWOD    128
#define NCLS    5

// LDS layout (floats):
//   sW1 : HID*TWOD        = 16384
//   sW2 : NCLS*HID        = 640
//   sB1 : HID             = 128
//   sB2 : 8               = 8
//   sM  : 8 waves * 16*128 = 16384   (per-wave m-tile, reused as h-tile)
#define SMEM_FLOATS (16384 + 640 + 128 + 8 + 8 * 16 * TWOD)

__global__ __launch_bounds__(256) void mlp_decoder_kernel(
    const float* __restrict__ ufeat, const float* __restrict__ ifeat,
    const float* __restrict__ W1, const float* __restrict__ b1,
    const float* __restrict__ W2, const float* __restrict__ b2,
    const long long* __restrict__ src_idx, const long long* __restrict__ dst_idx,
    float* __restrict__ out, int Etotal, int nTiles, int wavesTotal)
{
    extern __shared__ float smem[];
    float* sW1 = smem;                     // 16384
    float* sW2 = sW1 + HID * TWOD;         // 640
    float* sB1 = sW2 + NCLS * HID;         // 128
    float* sB2 = sB1 + HID;                // 8
    float* sM  = sB2 + 8;

    const int tid  = threadIdx.x;
    const int lane = tid & 31;
    const int wave = tid >> 5;
    const int half = lane >> 4;            // 0: lanes 0-15, 1: lanes 16-31
    const int l15  = lane & 15;
    const int c0   = half * 2;             // K sub-offset for A/B fragments

    float* mT = sM + wave * (16 * TWOD);   // this wave's 16x128 tile

    // ---- stage weights into LDS (once per block) ----
    for (int i = tid * 4; i < HID * TWOD; i += 256 * 4)
        *(v4f*)(sW1 + i) = *(const v4f*)(W1 + i);
    for (int i = tid; i < NCLS * HID; i += 256)
        sW2[i] = W2[i];
    if (tid < HID) sB1[tid] = b1[tid];
    if (tid < 8)   sB2[tid] = (tid < NCLS) ? b2[tid] : 0.0f;
    __syncthreads();

    const int gwave = blockIdx.x * 8 + wave;
    for (int tile = gwave; tile < nTiles; tile += wavesTotal) {
        const int ebase = tile * 16;

        // ---- gather indices: lanes 0-15 hold src, lanes 16-31 hold dst ----
        int eclamp = ebase + l15;
        if (eclamp >= Etotal) eclamp = Etotal - 1;
        const long long idx64 = (half == 0) ? src_idx[eclamp] : dst_idx[eclamp];
        const int idx32 = (int)idx64;      // NU/NM < 2^31

        // ---- cooperative gather of 16 rows into LDS m-tile ----
        // row r: lanes 0-15 fetch ufeat[src[r]][0..63], lanes 16-31 ifeat[dst[r]][0..63]
        #pragma unroll
        for (int r = 0; r < 16; ++r) {
            const int s  = __shfl(idx32, r);        // src index of row r
            const int dd = __shfl(idx32, r + 16);   // dst index of row r
            const float* rowp = (half == 0)
                ? (ufeat + (long long)s  * DFEAT + l15 * 4)
                : (ifeat + (long long)dd * DFEAT + l15 * 4);
            v4f v = *(const v4f*)rowp;
            *(v4f*)(mT + r * TWOD + half * DFEAT + l15 * 4) = v;
        }
        // same-wave DS ops are in-order: subsequent ds_loads see these stores

        // ---- load all A fragments for this tile (16x128 -> 32 K-steps) ----
        // A layout (16x4 f32): M = lane%16, VGPR{0,1} = K = 4k + c0 + {0,1}
        v2f a[32];
        #pragma unroll
        for (int k = 0; k < 32; ++k)
            a[k] = *(v2f*)(mT + l15 * TWOD + 4 * k + c0);

        // ---- layer 1: h[16x128] = m[16x128] @ W1^T, 8 N-tiles x 32 K-steps ----
        v8f acc[8] = {};
        #pragma unroll
        for (int n = 0; n < 8; ++n) {
            #pragma unroll
            for (int k = 0; k < 32; ++k) {
                // B layout (4x16): N = lane%16, VGPR{0,1} = K = 4k + c0 + {0,1}
                v2f bf = *(v2f*)(sW1 + (n * 16 + l15) * TWOD + 4 * k + c0);
                acc[n] = __builtin_amdgcn_wmma_f32_16x16x4_f32(
                    false, a[k], false, bf, (short)0, acc[n], false, false);
            }
        }

        // ---- bias + ReLU, write h back over the m-tile ----
        // C layout: VGPR i -> edge = i + 8*half, col N = n*16 + lane%16
        #pragma unroll
        for (int n = 0; n < 8; ++n) {
            const float bb = sB1[n * 16 + l15];
            #pragma unroll
            for (int i = 0; i < 8; ++i) {
                float h = acc[n][i] + bb;
                h = h > 0.0f ? h : 0.0f;
                mT[(i + half * 8) * TWOD + n * 16 + l15] = h;
            }
        }

        // ---- layer 2: out[16x5] = h @ W2^T + b2 (80 dot products over LDS) ----
        #pragma unroll
        for (int t = 0; t < 3; ++t) {
            const int o = t * 32 + lane;
            if (o < 16 * NCLS) {
                const int el = o / NCLS;
                const int c  = o % NCLS;
                const int eg = ebase + el;
                if (eg < Etotal) {
                    float sum = sB2[c];
                    const float* hrow = mT + el * TWOD;
                    const float* wrow = sW2 + c * HID;
                    #pragma unroll
                    for (int j = 0; j < HID; j += 4) {
                        v4f hv = *(v4f*)(hrow + j);
                        v4f wv = *(const v4f*)(wrow + j);
                        sum += hv.x * wv.x + hv.y * wv.y + hv.z * wv.z + hv.w * wv.w;
                    }
                    out[(long long)eg * NCLS + c] = sum;
                }
            }
        }
        // next iteration's gather stores are same-wave DS ops: ordered after reads
    }
}

extern "C" void kernel_launch(void* const* d_in, const int* in_sizes, int n_in,
                              void* d_out, int out_size, void* d_ws, size_t ws_size,
                              hipStream_t stream) {
    const float*     ufeat = (const float*)d_in[0];
    const float*     ifeat = (const float*)d_in[1];
    const float*     W1    = (const float*)d_in[2];
    const float*     b1    = (const float*)d_in[3];
    const float*     W2    = (const float*)d_in[4];
    const float*     b2    = (const float*)d_in[5];
    const long long* src   = (const long long*)d_in[6];
    const long long* dst   = (const long long*)d_in[7];
    float*           out   = (float*)d_out;

    const int E = in_sizes[6];            // 1,000,000 edges
    const int nTiles = (E + 15) / 16;     // 62,500 tiles of 16 edges

    const int blocks = 1024;              // grid-stride over tiles
    const int wavesTotal = blocks * 8;    // 256 threads = 8 wave32 per block
    const size_t shmem = (size_t)SMEM_FLOATS * sizeof(float);  // ~131 KB

    mlp_decoder_kernel<<<dim3(blocks), dim3(256), shmem, stream>>>(
        ufeat, ifeat, W1, b1, W2, b2, src, dst, out, E, nTiles, wavesTotal);
}